// EdgePredictor_31662498906597
// MI455X (gfx1250) — compile-verified
//
#include <hip/hip_runtime.h>
#include <hip/hip_bf16.h>

typedef __attribute__((ext_vector_type(2))) float v2f;
typedef __attribute__((ext_vector_type(8))) float v8f;

#define D 128         // IN_FEATS
#define KSTEPS (D/4)  // 32 WMMA K-steps of K=4

// ---------------------------------------------------------------------------
// Pass 1: per-node partial scores via V_WMMA_F32_16X16X4_F32 (fp32 WMMA).
//   s_src[n] = dot(h[n,:], W[0:128])     (column N=0 of the 16x16 C tile)
//   s_dst[n] = dot(h[n,:], W[128:256])   (column N=1)
// One wave handles a 16-node tile; K=128 consumed in 32 steps of K=4.
// A layout (16x4 f32): lane L -> M = L%16; VGPR0: K = 2*(L/16), VGPR1: K+1.
// B layout (4x16 f32): lane L -> N = L%16; VGPR0: K = 2*(L/16), VGPR1: K+1.
// C layout (16x16 f32): lane L -> N = L%16; VGPR v -> M = v + 8*(L/16).
// B is loop-invariant: preloaded branch-free into 32 v2f fragments so the
// hot loop is pure {global_load_b64 A, v_wmma} with no exec juggling.
// ---------------------------------------------------------------------------
__global__ __launch_bounds__(256) void node_scores_wmma(
    const float* __restrict__ h, const float* __restrict__ W,
    float* __restrict__ s_src, float* __restrict__ s_dst,
    int n_nodes, int n_tiles)
{
  const int wave = threadIdx.x >> 5;
  const int lane = threadIdx.x & 31;
  const int tile = blockIdx.x * (blockDim.x >> 5) + wave;
  if (tile >= n_tiles) return;            // wave-uniform: EXEC stays full

  const int n    = lane & 15;             // A: M index / B: N index
  const int grp  = lane >> 4;             // half-wave group
  const int koff = grp << 1;              // K offset inside each 4-wide step

  // ---- Preload B fragments (branch-free: clamp address, mask value) ----
  const int   sel = (n < 2) ? n : 0;      // safe address for inactive columns
  const float msk = (n < 2) ? 1.f : 0.f;  // zero out columns N >= 2
  const float* __restrict__ wp = W + sel * D + koff;
  v2f bfrag[KSTEPS];
#pragma unroll
  for (int s = 0; s < KSTEPS; ++s) {
    v2f wv = *(const v2f*)(wp + 4 * s);   // 8B-aligned (koff even)
    bfrag[s].x = wv.x * msk;
    bfrag[s].y = wv.y * msk;
  }

  // ---- A row pointer (clamped so EXEC stays all-1s for WMMA) ----
  int row = tile * 16 + n;
  if (row > n_nodes - 1) row = n_nodes - 1;
  const float* __restrict__ arow = h + (size_t)row * D + koff;

  v8f acc = {0.f, 0.f, 0.f, 0.f, 0.f, 0.f, 0.f, 0.f};
#pragma unroll
  for (int s = 0; s < KSTEPS; ++s) {
    v2f a = *(const v2f*)(arow + 4 * s);
    acc = __builtin_amdgcn_wmma_f32_16x16x4_f32(
        /*neg_a=*/false, a, /*neg_b=*/false, bfrag[s],
        /*c_mod=*/(short)0, acc, /*reuse_a=*/false, /*reuse_b=*/false);
  }

  if (n < 2) {
    float* __restrict__ outp = (n == 0) ? s_src : s_dst;
    const int baseM = tile * 16 + grp * 8;
#pragma unroll
    for (int v = 0; v < 8; ++v) {
      int r = baseM + v;
      if (r < n_nodes) outp[r] = acc[v];
    }
  }
}

// ---------------------------------------------------------------------------
// Pass 2: per-edge gather of two scalars from the L2-resident score table.
// ---------------------------------------------------------------------------
__global__ __launch_bounds__(256) void edge_gather(
    const float* __restrict__ s_src, const float* __restrict__ s_dst,
    const int* __restrict__ src_idx, const int* __restrict__ dst_idx,
    const float* __restrict__ bias, float* __restrict__ out, int n_edges)
{
  int e = blockIdx.x * blockDim.x + threadIdx.x;
  if (e >= n_edges) return;
  out[e] = s_src[src_idx[e]] + s_dst[dst_idx[e]] + bias[0];
}

// ---------------------------------------------------------------------------
// Fallback (workspace too small): direct per-edge dot products.
// ---------------------------------------------------------------------------
__global__ __launch_bounds__(256) void edge_naive(
    const float* __restrict__ h, const int* __restrict__ src_idx,
    const int* __restrict__ dst_idx, const float* __restrict__ W,
    const float* __restrict__ bias, float* __restrict__ out, int n_edges)
{
  int e = blockIdx.x * blockDim.x + threadIdx.x;
  if (e >= n_edges) return;
  const float* hs = h + (size_t)src_idx[e] * D;
  const float* hd = h + (size_t)dst_idx[e] * D;
  float acc = bias[0];
#pragma unroll 8
  for (int i = 0; i < D; ++i) {
    acc += hs[i] * W[i] + hd[i] * W[D + i];
  }
  out[e] = acc;
}

extern "C" void kernel_launch(void* const* d_in, const int* in_sizes, int n_in,
                              void* d_out, int out_size, void* d_ws, size_t ws_size,
                              hipStream_t stream) {
  const float* h      = (const float*)d_in[0];   // [N, 128]
  const int*   srcI   = (const int*)  d_in[1];   // [E]
  const int*   dstI   = (const int*)  d_in[2];   // [E]
  const float* W      = (const float*)d_in[3];   // [1, 256]
  const float* bias   = (const float*)d_in[4];   // [1]
  float*       out    = (float*)d_out;           // [E, 1]

  const int n_nodes = in_sizes[0] / D;
  const int n_edges = in_sizes[1];

  const size_t need = 2 * (size_t)n_nodes * sizeof(float);
  if (ws_size >= need) {
    float* s_src = (float*)d_ws;
    float* s_dst = s_src + n_nodes;

    const int n_tiles = (n_nodes + 15) / 16;
    const int waves_per_block = 8;                 // 256 threads = 8 wave32
    const int blocks1 = (n_tiles + waves_per_block - 1) / waves_per_block;
    node_scores_wmma<<<blocks1, 256, 0, stream>>>(h, W, s_src, s_dst,
                                                  n_nodes, n_tiles);

    const int blocks2 = (n_edges + 255) / 256;
    edge_gather<<<blocks2, 256, 0, stream>>>(s_src, s_dst, srcI, dstI,
                                             bias, out, n_edges);
  } else {
    const int blocks = (n_edges + 255) / 256;
    edge_naive<<<blocks, 256, 0, stream>>>(h, srcI, dstI, W, bias, out, n_edges);
  }
}